// GraphEncoder_23596550324443
// MI455X (gfx1250) — compile-verified
//
#include <hip/hip_runtime.h>

// ---------------------------------------------------------------------------
// GraphEncoder on MI455X (gfx1250, wave32).
// fp32 end-to-end (matches reference). Matrix ops: V_WMMA_F32_16X16X4_F32.
// ---------------------------------------------------------------------------

typedef float v2f __attribute__((ext_vector_type(2)));
typedef float v8f __attribute__((ext_vector_type(8)));

#define NPTS  4096
#define BATCH 8
#define KNN   16
#define CHUNK 512          // columns of the distance matrix per LDS pass
#define DPAD  516          // row pitch (floats) -> bank-conflict-free LDS

// ---------------------------------------------------------------------------
// 1) per-point squared coordinate norms
// ---------------------------------------------------------------------------
__global__ void __launch_bounds__(256) sqnorm_kernel(const float* __restrict__ feats,
                                                     float* __restrict__ sq) {
  int p = blockIdx.x * 256 + threadIdx.x;          // 0 .. 32767
  const float* f = feats + (size_t)p * 6;
  float x = f[0], y = f[1], z = f[2];
  sq[p] = x * x + y * y + z * z;
}

// ---------------------------------------------------------------------------
// 2) kNN. Block = 16 rows of one batch; distances via f32 WMMA (K=3 pad 4)
//    streamed into LDS in 512-col chunks; top-16 via register insertion.
//    Operand loads are branchless: floats {ka, ka+1} (ka in {0,2}) are always
//    in-bounds of the 6-float row, so one aligned b64 load + cndmask pad.
// ---------------------------------------------------------------------------
__global__ void __launch_bounds__(256) knn_kernel(const float* __restrict__ feats,
                                                  const float* __restrict__ sq,
                                                  int* __restrict__ nbr) {
  __shared__ float dist[16][DPAD];    // ~33 KB
  __shared__ float sqr_sh[16];

  int b       = blockIdx.x >> 8;        // 256 row-tiles per batch
  int r0      = (blockIdx.x & 255) << 4;
  int lane    = threadIdx.x & 31;
  int wave    = threadIdx.x >> 5;       // 0..7
  int laneRow = lane & 15;
  int laneHi  = lane >> 4;              // 0 or 1

  const float* fb  = feats + (size_t)b * NPTS * 6;
  const float* sqb = sq + b * NPTS;

  if (threadIdx.x < 16) sqr_sh[threadIdx.x] = sqb[r0 + threadIdx.x];
  __syncthreads();

  // A operand (16x4 f32): lanes 0-15 hold K=0,1 ; lanes 16-31 hold K=2,3
  int ka   = laneHi * 2;
  int arow = r0 + laneRow;
  v2f a = *(const v2f*)(fb + arow * 6 + ka);   // aligned b64, in-bounds
  a[1] = laneHi ? 0.0f : a[1];                 // zero K=3 pad (cndmask)

  // running top-16 (ascending); only threads 0..15 use them
  float best[16]; int bidx[16];
#pragma unroll
  for (int q = 0; q < 16; ++q) { best[q] = 1e38f; bidx[q] = 0; }

  for (int chunk = 0; chunk < NPTS / CHUNK; ++chunk) {
    int cbase = chunk * CHUNK;

    // ---- compute phase: 8 waves x 4 tiles = 512 columns ----
#pragma unroll
    for (int t = 0; t < 4; ++t) {
      int c0  = cbase + ((wave << 2) + t) * 16;
      int col = c0 + laneRow;

      v2f bv = *(const v2f*)(fb + col * 6 + ka);  // aligned b64, in-bounds
      bv[1] = laneHi ? 0.0f : bv[1];
      float sqc = sqb[col];

      v8f acc = {0.f, 0.f, 0.f, 0.f, 0.f, 0.f, 0.f, 0.f};
      acc = __builtin_amdgcn_wmma_f32_16x16x4_f32(false, a, false, bv,
                                                  (short)0, acc, false, false);
      bool diagTile = (c0 == r0);                 // only 1 of 256 tiles
#pragma unroll
      for (int v = 0; v < 8; ++v) {
        int   m  = v + (laneHi << 3);             // output row within strip
        float d2 = sqr_sh[m] + sqc - 2.0f * acc[v];
        if (diagTile && laneRow == m) d2 = 1e30f; // mask self (diagonal)
        dist[m][(c0 - cbase) + laneRow] = d2;
      }
    }
    __syncthreads();

    // ---- selection phase: thread t owns row t ----
    if (threadIdx.x < 16) {
      int t = threadIdx.x;
      for (int j = 0; j < CHUNK; ++j) {
        float d = dist[t][j];
        if (d < best[15]) {
          float v = d; int vi = cbase + j;
#pragma unroll
          for (int q = 0; q < 16; ++q) {
            if (v < best[q]) {
              float tb = best[q]; int ti = bidx[q];
              best[q] = v; bidx[q] = vi; v = tb; vi = ti;
            }
          }
        }
      }
    }
    __syncthreads();
  }

  if (threadIdx.x < 16) {
    int* out = nbr + ((size_t)(b * NPTS + r0 + threadIdx.x)) * KNN;
#pragma unroll
    for (int q = 0; q < 16; ++q) out[q] = bidx[q];
  }
}

// ---------------------------------------------------------------------------
// 3) fp32 WMMA GEMM, compile-time K/N, 4 M-tiles per wave (B-fragment reuse).
//    C[M,N] = A[M,K] @ B[K,N] (+ bias). K%4==0 path: zero guards, b64 A-loads.
//    K=6 path: clamped in-bounds addresses + branchless value selects.
// ---------------------------------------------------------------------------
template <int KD, int NW, bool BIAS>
__global__ void __launch_bounds__(256) gemm_wmma_f32(const float* __restrict__ A,
                                                     const float* __restrict__ Bm,
                                                     const float* __restrict__ bias,
                                                     float* __restrict__ C) {
  constexpr int MT    = 4;                 // M-tiles per wave
  constexpr int KPAD  = (KD + 3) & ~3;
  constexpr int tilesN = NW / 16;

  int lane    = threadIdx.x & 31;
  int wave    = threadIdx.x >> 5;
  int laneRow = lane & 15;
  int laneHi  = lane >> 4;

  int wid  = blockIdx.x * 8 + wave;
  int tm0  = (wid / tilesN) * MT;          // first of 4 consecutive M-tiles
  int tn   = wid % tilesN;
  int col  = tn * 16 + laneRow;

  v8f acc[MT];
#pragma unroll
  for (int m = 0; m < MT; ++m) acc[m] = (v8f){0.f,0.f,0.f,0.f,0.f,0.f,0.f,0.f};

#pragma unroll
  for (int k0 = 0; k0 < KPAD; k0 += 4) {
    int k = k0 + laneHi * 2;

    // B fragment: rows k, k+1 at column col (reused by all MT WMMAs)
    v2f bv;
    if constexpr (KD % 4 == 0) {
      bv[0] = Bm[(size_t)k * NW + col];
      bv[1] = Bm[(size_t)(k + 1) * NW + col];
    } else {
      int k0c = (k     < KD) ? k     : 0;   // clamp -> always in-bounds
      int k1c = (k + 1 < KD) ? k + 1 : 0;
      float b0 = Bm[(size_t)k0c * NW + col];
      float b1 = Bm[(size_t)k1c * NW + col];
      bv[0] = (k     < KD) ? b0 : 0.0f;
      bv[1] = (k + 1 < KD) ? b1 : 0.0f;
    }

#pragma unroll
    for (int m = 0; m < MT; ++m) {
      int row = (tm0 + m) * 16 + laneRow;
      v2f av;
      if constexpr (KD % 4 == 0) {
        av = *(const v2f*)(A + (size_t)row * KD + k);          // aligned b64
      } else {
        int kc = (k < KD) ? k : 0;                             // clamp
        av = *(const v2f*)(A + (size_t)row * KD + kc);         // in-bounds
        av[0] = (k     < KD) ? av[0] : 0.0f;
        av[1] = (k + 1 < KD) ? av[1] : 0.0f;
      }
      acc[m] = __builtin_amdgcn_wmma_f32_16x16x4_f32(false, av, false, bv,
                                                     (short)0, acc[m],
                                                     false, false);
    }
  }

  float bval = 0.0f;
  if constexpr (BIAS) bval = bias[col];
#pragma unroll
  for (int m = 0; m < MT; ++m) {
#pragma unroll
    for (int v = 0; v < 8; ++v) {
      int r = (tm0 + m) * 16 + v + laneHi * 8;
      C[(size_t)r * NW + col] = acc[m][v] + bval;
    }
  }
}

// ---------------------------------------------------------------------------
// 4) neighbor aggregation: x = relu((sum_{j in nbr} h[j] + h[i]) / 17 + b)
//    Wave per point; lane owns PER channels. Indices loaded up-front and all
//    16 gathered rows prefetched (global_prefetch_b8) before the accumulate.
// ---------------------------------------------------------------------------
template <int PER>
__global__ void __launch_bounds__(256) aggregate_kernel(const float* __restrict__ h,
                                                        const int* __restrict__ nbr,
                                                        const float* __restrict__ bias,
                                                        float* __restrict__ xout) {
  const int Cch = PER * 32;
  int lane = threadIdx.x & 31;
  int wave = threadIdx.x >> 5;
  int p    = blockIdx.x * 8 + wave;     // global point 0..32767
  int b    = p >> 12;                   // batch index

  const float* hp = h + (size_t)p * Cch;
  float acc[PER];
#pragma unroll
  for (int i = 0; i < PER; ++i) acc[i] = hp[lane * PER + i];

  const int*   np = nbr + (size_t)p * KNN;
  const float* hb = h + (size_t)b * NPTS * Cch;

  int idx[KNN];
#pragma unroll
  for (int j = 0; j < KNN; ++j) idx[j] = np[j];
#pragma unroll
  for (int j = 0; j < KNN; ++j)
    __builtin_prefetch(hb + (size_t)idx[j] * Cch + lane * PER, 0, 3);

#pragma unroll
  for (int j = 0; j < KNN; ++j) {
    const float* hn = hb + (size_t)idx[j] * Cch;
#pragma unroll
    for (int i = 0; i < PER; ++i) acc[i] += hn[lane * PER + i];
  }

  const float inv_deg = 1.0f / (KNN + 1);
#pragma unroll
  for (int i = 0; i < PER; ++i) {
    float v = acc[i] * inv_deg + bias[lane * PER + i];
    xout[(size_t)p * Cch + lane * PER + i] = fmaxf(v, 0.0f);
  }
}

// ---------------------------------------------------------------------------
extern "C" void kernel_launch(void* const* d_in, const int* in_sizes, int n_in,
                              void* d_out, int out_size, void* d_ws, size_t ws_size,
                              hipStream_t stream) {
  (void)in_sizes; (void)n_in; (void)out_size; (void)ws_size;
  const float* feats = (const float*)d_in[0];   // (8,4096,6)
  const float* W1    = (const float*)d_in[1];   // (6,64)
  const float* b1    = (const float*)d_in[2];   // (64,)
  const float* W2    = (const float*)d_in[3];   // (64,128)
  const float* b2    = (const float*)d_in[4];   // (128,)
  const float* Wf    = (const float*)d_in[5];   // (128,128)
  const float* bf    = (const float*)d_in[6];   // (128,)
  // d_in[7] = k (16), hardcoded per reference

  char*  ws  = (char*)d_ws;
  float* sq  = (float*)(ws);                        // 128 KB
  int*   nbr = (int*)  (ws + ((size_t)1  << 20));   // 2 MB
  float* h1  = (float*)(ws + ((size_t)4  << 20));   // 8 MB
  float* x1  = (float*)(ws + ((size_t)12 << 20));   // 8 MB
  float* h2  = (float*)(ws + ((size_t)20 << 20));   // 16 MB
  float* x2  = (float*)(ws + ((size_t)36 << 20));   // 16 MB

  sqnorm_kernel<<<128, 256, 0, stream>>>(feats, sq);
  knn_kernel<<<BATCH * (NPTS / 16), 256, 0, stream>>>(feats, sq, nbr);

  // M = 32768 rows; waves = (M/16/4) * (N/16); 8 waves per block
  // layer 1: h1 = feats @ W1 ; x1 = relu((gather-sum + h1)/17 + b1)
  gemm_wmma_f32<6, 64, false><<<(512 * 4) / 8, 256, 0, stream>>>(feats, W1, nullptr, h1);
  aggregate_kernel<2><<<32768 / 8, 256, 0, stream>>>(h1, nbr, b1, x1);

  // layer 2: h2 = x1 @ W2 ; x2 = relu((gather-sum + h2)/17 + b2)
  gemm_wmma_f32<64, 128, false><<<(512 * 8) / 8, 256, 0, stream>>>(x1, W2, nullptr, h2);
  aggregate_kernel<4><<<32768 / 8, 256, 0, stream>>>(h2, nbr, b2, x2);

  // output: out = x2 @ Wf + bf
  gemm_wmma_f32<128, 128, true><<<(512 * 8) / 8, 256, 0, stream>>>(x2, Wf, bf, (float*)d_out);
}